// GCN_18476949307699
// MI455X (gfx1250) — compile-verified
//
#include <hip/hip_runtime.h>
#include <hip/hip_bf16.h>
#include <math.h>

typedef __attribute__((ext_vector_type(2))) float v2f;
typedef __attribute__((ext_vector_type(8))) float v8f;

// ---------------- utility kernels ----------------

__global__ void gcn_zero_kernel(float* __restrict__ p, int n) {
    int i = blockIdx.x * blockDim.x + threadIdx.x;
    if (i < n) p[i] = 0.0f;
}

__global__ void gcn_degree_kernel(const int* __restrict__ src, const int* __restrict__ dst,
                                  float* __restrict__ outdeg, float* __restrict__ indeg, int n_edges) {
    int i = blockIdx.x * blockDim.x + threadIdx.x;
    if (i < n_edges) {
        atomicAdd(&outdeg[src[i]], 1.0f);
        atomicAdd(&indeg[dst[i]], 1.0f);
    }
}

__global__ void gcn_norm_kernel(float* __restrict__ p, int n) {
    int i = blockIdx.x * blockDim.x + threadIdx.x;
    if (i < n) {
        float d = p[i];
        d = (d < 1.0f) ? 1.0f : d;
        p[i] = 1.0f / sqrtf(d);
    }
}

// ---------------- WMMA GEMM: out[n, NT*16] = (X[n,:K] * scale[n]) @ W[K, NT*16] ----------------
// One wave per 16-row node tile; fp32 WMMA 16x16x4, K-loop of 4, NT column tiles held in regs.
// A layout (f32 16x4): lane L<16 holds row M=L, K = {0,1}; lane L>=16 holds row M=L-16, K = {2,3}.
// B layout (f32 4x16): lane L<16 holds col N=L, K = {0,1}; lane L>=16 holds col N=L-16, K = {2,3}.
// C/D layout: VGPR v, lane L: row = v + 8*(L>=16), col = L&15.

template <int NT>
__global__ __launch_bounds__(256)
void gcn_gemm_scaled_wmma(const float* __restrict__ X, const float* __restrict__ scale,
                          const float* __restrict__ W, float* __restrict__ out,
                          int n_nodes, int K) {
    const int NOUT = NT * 16;
    int wave = (blockIdx.x * blockDim.x + threadIdx.x) >> 5;
    int lane = threadIdx.x & 31;
    int row0 = wave * 16;
    if (row0 >= n_nodes) return;   // uniform per wave -> whole wave exits, EXEC stays all-1 for WMMA

    int hi   = lane >> 4;          // 0 or 1
    int kOff = hi * 2;             // K sub-offset for both A and B halves
    int mrow = row0 + (lane & 15);
    int col  = lane & 15;

    float s = scale[mrow];
    const float* __restrict__ xrow = X + (size_t)mrow * (size_t)K;

    v8f acc[NT];
    v8f vzero = {};
#pragma unroll
    for (int nt = 0; nt < NT; ++nt) acc[nt] = vzero;

    for (int k0 = 0; k0 < K; k0 += 4) {
        float2 av = *(const float2*)(xrow + k0 + kOff);   // 8B-aligned: k0%4==0, kOff in {0,2}
        v2f a;
        a.x = av.x * s;
        a.y = av.y * s;
        const float* __restrict__ wr = W + (size_t)(k0 + kOff) * (size_t)NOUT;
#pragma unroll
        for (int nt = 0; nt < NT; ++nt) {
            v2f b;
            b.x = wr[nt * 16 + col];
            b.y = wr[NOUT + nt * 16 + col];
            acc[nt] = __builtin_amdgcn_wmma_f32_16x16x4_f32(
                /*neg_a=*/false, a, /*neg_b=*/false, b,
                /*c_mod=*/(short)0, acc[nt], /*reuse_a=*/false, /*reuse_b=*/false);
        }
    }

    int crow = row0 + hi * 8;
#pragma unroll
    for (int nt = 0; nt < NT; ++nt) {
#pragma unroll
        for (int v = 0; v < 8; ++v) {
            out[(size_t)(crow + v) * (size_t)NOUT + nt * 16 + col] = acc[nt][v];
        }
    }
}

// ---------------- edge scatter-add: agg[dst[e], :F] += hw[src[e], :F] ----------------
// F/4 threads per edge, float4 per thread: one wave covers one (F=128) edge, 512B contiguous.

template <int F>
__global__ void gcn_scatter_add_kernel(const float* __restrict__ hw, const int* __restrict__ src,
                                       const int* __restrict__ dst, float* __restrict__ agg,
                                       int n_edges) {
    constexpr int TPE = F / 4;
    int gid = blockIdx.x * blockDim.x + threadIdx.x;
    int e = gid / TPE;
    if (e >= n_edges) return;
    int t = gid % TPE;
    int s = src[e];
    int d = dst[e];
    const float4 v = *(const float4*)(hw + (size_t)s * F + t * 4);
    float* o = agg + (size_t)d * F + t * 4;
    atomicAdd(o + 0, v.x);
    atomicAdd(o + 1, v.y);
    atomicAdd(o + 2, v.z);
    atomicAdd(o + 3, v.w);
}

// ---------------- finalize: out = act(agg * dst_norm[node] + bias[f]) ----------------

__global__ void gcn_finalize_kernel(const float* __restrict__ agg, const float* __restrict__ dnorm,
                                    const float* __restrict__ bias, float* __restrict__ out,
                                    int n_nodes, int F, int act) {
    int i = blockIdx.x * blockDim.x + threadIdx.x;
    if (i >= n_nodes * F) return;
    int node = i / F;
    int f = i % F;
    float v = agg[i] * dnorm[node] + bias[f];
    if (act == 0) {
        v = (v > 0.0f) ? v : 0.0f;          // relu
    } else {
        v = 1.0f / (1.0f + expf(-v)) + 1e-8f;  // sigmoid + 1e-8
    }
    out[i] = v;
}

// ---------------- driver ----------------

extern "C" void kernel_launch(void* const* d_in, const int* in_sizes, int n_in,
                              void* d_out, int out_size, void* d_ws, size_t ws_size,
                              hipStream_t stream) {
    (void)n_in; (void)out_size; (void)ws_size;

    const float* feat = (const float*)d_in[0];
    const int*   src  = (const int*)d_in[1];
    const int*   dst  = (const int*)d_in[2];
    const float* W0   = (const float*)d_in[3];
    const float* b0   = (const float*)d_in[4];
    const float* W1   = (const float*)d_in[5];
    const float* b1   = (const float*)d_in[6];
    const float* W2   = (const float*)d_in[7];
    const float* b2   = (const float*)d_in[8];

    const int F = 128;                 // IN_FEATS == N_HIDDEN
    const int C = 16;                  // N_CLASSES
    const int N = in_sizes[0] / F;     // 100000
    const int E = in_sizes[1];         // 1600000

    // workspace layout (floats): snorm[N] | dnorm[N] | bufA[N*F] | bufB[N*F]
    float* snorm = (float*)d_ws;
    float* dnorm = snorm + N;
    float* bufA  = dnorm + N;
    float* bufB  = bufA + (size_t)N * F;

    const int TB = 256;
    auto nblk = [](long long n, int tb) { return (unsigned)((n + tb - 1) / tb); };

    // degrees + norms
    gcn_zero_kernel<<<nblk(2LL * N, TB), TB, 0, stream>>>(snorm, 2 * N);
    gcn_degree_kernel<<<nblk(E, TB), TB, 0, stream>>>(src, dst, snorm, dnorm, E);
    gcn_norm_kernel<<<nblk(2LL * N, TB), TB, 0, stream>>>(snorm, 2 * N);

    const long long gemm_threads = (long long)((N + 15) / 16) * 32;
    const long long nf = (long long)N * F;
    const long long nc = (long long)N * C;

    // ---- layer 0: feat -> bufA(hw) -> bufB(agg) -> bufA(h1) ----
    gcn_gemm_scaled_wmma<8><<<nblk(gemm_threads, TB), TB, 0, stream>>>(feat, snorm, W0, bufA, N, F);
    gcn_zero_kernel<<<nblk(nf, TB), TB, 0, stream>>>(bufB, (int)nf);
    gcn_scatter_add_kernel<128><<<nblk((long long)E * 32, TB), TB, 0, stream>>>(bufA, src, dst, bufB, E);
    gcn_finalize_kernel<<<nblk(nf, TB), TB, 0, stream>>>(bufB, dnorm, b0, bufA, N, F, 0);

    // ---- layer 1: bufA(h1) -> bufB(hw) -> bufA(agg) -> bufB(h2) ----
    gcn_gemm_scaled_wmma<8><<<nblk(gemm_threads, TB), TB, 0, stream>>>(bufA, snorm, W1, bufB, N, F);
    gcn_zero_kernel<<<nblk(nf, TB), TB, 0, stream>>>(bufA, (int)nf);
    gcn_scatter_add_kernel<128><<<nblk((long long)E * 32, TB), TB, 0, stream>>>(bufB, src, dst, bufA, E);
    gcn_finalize_kernel<<<nblk(nf, TB), TB, 0, stream>>>(bufA, dnorm, b1, bufB, N, F, 0);

    // ---- layer 2 (F_out = 16): bufB(h2) -> bufA(hw) -> bufB(agg) -> d_out ----
    gcn_gemm_scaled_wmma<1><<<nblk(gemm_threads, TB), TB, 0, stream>>>(bufB, snorm, W2, bufA, N, F);
    gcn_zero_kernel<<<nblk(nc, TB), TB, 0, stream>>>(bufB, (int)nc);
    gcn_scatter_add_kernel<16><<<nblk((long long)E * 4, TB), TB, 0, stream>>>(bufA, src, dst, bufB, E);
    gcn_finalize_kernel<<<nblk(nc, TB), TB, 0, stream>>>(bufB, dnorm, b2, (float*)d_out, N, C, 1);
}